// RNNAugmentCell_27633819582750
// MI455X (gfx1250) — compile-verified
//
#include <hip/hip_runtime.h>
#include <cstdint>

// ---------------- gfx1250 feature probes (device pass only) ----------------
#define AS1 __attribute__((address_space(1)))
#define AS3 __attribute__((address_space(3)))

typedef int v4i __attribute__((vector_size(16)));

#if defined(__AMDGCN__) && __has_builtin(__builtin_amdgcn_global_load_async_to_lds_b128)
#define HAVE_ASYNC 1
#else
#define HAVE_ASYNC 0
#endif

#if defined(__AMDGCN__) && __has_builtin(__builtin_amdgcn_s_wait_asynccnt)
#define WAIT_ASYNC(n) __builtin_amdgcn_s_wait_asynccnt(n)
#else
#define WAIT_ASYNC(n) asm volatile("s_wait_asynccnt %0" ::"i"(n) : "memory")
#endif

// per-lane 16B async copy from global address gp to LDS address lp; tracked by ASYNCcnt.
// Builtin signature (from hipcc diagnostic): (global int4*, shared int4*, imm, imm)
#define GASYNC16(gp, lp)                                          \
  __builtin_amdgcn_global_load_async_to_lds_b128(                 \
      (AS1 v4i*)(uintptr_t)(gp),                                  \
      (AS3 v4i*)(uint32_t)(uintptr_t)(lp), 0, 0)

// ---------------- fast activations (hw trans ops) ----------------
__device__ __forceinline__ float fast_exp2(float x) {
#if defined(__AMDGCN__) && __has_builtin(__builtin_amdgcn_exp2f)
  return __builtin_amdgcn_exp2f(x);
#else
  return exp2f(x);
#endif
}
__device__ __forceinline__ float fast_rcp(float x) {
#if defined(__AMDGCN__) && __has_builtin(__builtin_amdgcn_rcpf)
  return __builtin_amdgcn_rcpf(x);
#else
  return 1.0f / x;
#endif
}
__device__ __forceinline__ float fast_tanh(float x) {
#if defined(__AMDGCN__) && __has_builtin(__builtin_amdgcn_tanhf)
  return __builtin_amdgcn_tanhf(x);  // CDNA5 V_TANH_F32, single trans op
#else
  // tanh(x) = 1 - 2/(exp(2x)+1)
  float e = fast_exp2(x * 2.8853900817779268f);  // 2*log2(e)
  return 1.0f - 2.0f * fast_rcp(e + 1.0f);
#endif
}
__device__ __forceinline__ float fast_sigmoid(float x) {
  // 1/(1+exp(-x)) = 1/(1+exp2(-x*log2(e)))
  return fast_rcp(1.0f + fast_exp2(x * -1.4426950408889634f));
}

// ---------------- problem geometry ----------------
constexpr int TLEN = 2048;   // T
constexpr int NROWS = 8192;  // B*C = 32*256
constexpr int RPB = 8;       // rows per block
constexpr int POOLN = 128;   // pooled length
// output channel-block stride = 256*128
constexpr int CH_BLK = 256 * POOLN;

// Each thread processes half a pooling window (8 elems), computes the DAG
// (sharing tanh(s0) and sigmoid-family values across nodes), accumulates the
// four node sums, pair-reduces via shfl_xor, even lanes store 4 outputs.
__device__ __forceinline__ void compute_half_window(
    const float* __restrict__ a, const float* __restrict__ b, int row, int w,
    int h, float* __restrict__ out) {
  float4 A0 = *(const float4*)(a);
  float4 A1 = *(const float4*)(a + 4);
  float4 B0 = *(const float4*)(b);
  float4 B1 = *(const float4*)(b + 4);
  float xa[8] = {A0.x, A0.y, A0.z, A0.w, A1.x, A1.y, A1.z, A1.w};
  float xb[8] = {B0.x, B0.y, B0.z, B0.w, B1.x, B1.y, B1.z, B1.w};
  float acc2 = 0.f, acc3 = 0.f, acc4 = 0.f, acc5 = 0.f;
#pragma unroll
  for (int k = 0; k < 8; ++k) {
    float t0 = fast_tanh(xa[k]);                 // shared by s2 and s4
    float v2 = t0 + fmaxf(xb[k], 0.f);           // s2 = tanh(s0)+relu(s1)
    float v3 = fast_sigmoid(xb[k]) + v2;         // s3 = sig(s1)+s2
    float v4 = t0 + fmaxf(v3, 0.f);              // s4 = tanh(s0)+relu(s3)
    float v5 = fast_sigmoid(v2) + v4;            // s5 = sig(s2)+s4
    acc2 += v2;
    acc3 += v3;
    acc4 += v4;
    acc5 += v5;
  }
  acc2 += __shfl_xor(acc2, 1);
  acc3 += __shfl_xor(acc3, 1);
  acc4 += __shfl_xor(acc4, 1);
  acc5 += __shfl_xor(acc5, 1);
  if (h == 0) {
    const int bb = row >> 8;   // batch
    const int cc = row & 255;  // channel within C
    size_t base = (((size_t)bb * 1024 + cc) * POOLN) + w;
    const float s = 1.0f / 16.0f;
    out[base + 0 * CH_BLK] = acc2 * s;
    out[base + 1 * CH_BLK] = acc3 * s;
    out[base + 2 * CH_BLK] = acc4 * s;
    out[base + 3 * CH_BLK] = acc5 * s;
  }
}

__global__ __launch_bounds__(256) void darts_cell_pool(
    const float* __restrict__ s0, const float* __restrict__ s1,
    float* __restrict__ out) {
  const int tid = threadIdx.x;
  const int w = tid >> 1;  // pooling window 0..127
  const int h = tid & 1;   // half-window id
  const int row0 = blockIdx.x * RPB;

#if HAVE_ASYNC
  // double-buffered async staging: 2 x (8KB s0-row + 8KB s1-row) = 32KB LDS
  __shared__ float lds[2][2 * TLEN];
  {  // prologue: prefetch row0 into buffer 0 (4 x 4KB async issues)
    const char* g0 = (const char*)(s0 + (size_t)row0 * TLEN);
    const char* g1 = (const char*)(s1 + (size_t)row0 * TLEN);
    char* l = (char*)&lds[0][0];
    GASYNC16(g0 + tid * 16, l + tid * 16);
    GASYNC16(g0 + 4096 + tid * 16, l + 4096 + tid * 16);
    GASYNC16(g1 + tid * 16, l + 8192 + tid * 16);
    GASYNC16(g1 + 4096 + tid * 16, l + 12288 + tid * 16);
  }
  for (int i = 0; i < RPB; ++i) {
    const int row = row0 + i;
    const int buf = i & 1;
    if (i + 1 < RPB) {  // issue next row's copies, then drain current buffer
      const char* g0 = (const char*)(s0 + (size_t)(row + 1) * TLEN);
      const char* g1 = (const char*)(s1 + (size_t)(row + 1) * TLEN);
      char* l = (char*)&lds[buf ^ 1][0];
      GASYNC16(g0 + tid * 16, l + tid * 16);
      GASYNC16(g0 + 4096 + tid * 16, l + 4096 + tid * 16);
      GASYNC16(g1 + tid * 16, l + 8192 + tid * 16);
      GASYNC16(g1 + 4096 + tid * 16, l + 12288 + tid * 16);
      WAIT_ASYNC(4);  // <=4 outstanding: the 4 newest (next buffer)
    } else {
      WAIT_ASYNC(0);
    }
    __syncthreads();  // all lanes' copies into buf visible
    compute_half_window(&lds[buf][w * 16 + h * 8],
                        &lds[buf][TLEN + w * 16 + h * 8], row, w, h, out);
    __syncthreads();  // done reading buf before it is overwritten
  }
#else
  // fallback: direct vectorized global loads (still single-pass over inputs)
  for (int i = 0; i < RPB; ++i) {
    const int row = row0 + i;
    const size_t off = (size_t)row * TLEN + (size_t)w * 16 + h * 8;
    compute_half_window(s0 + off, s1 + off, row, w, h, out);
  }
#endif
}

extern "C" void kernel_launch(void* const* d_in, const int* in_sizes, int n_in,
                              void* d_out, int out_size, void* d_ws,
                              size_t ws_size, hipStream_t stream) {
  (void)in_sizes;
  (void)n_in;
  (void)out_size;
  (void)d_ws;
  (void)ws_size;
  const float* s0 = (const float*)d_in[0];
  const float* s1 = (const float*)d_in[1];
  float* out = (float*)d_out;
  dim3 grid(NROWS / RPB);  // 1024 blocks
  dim3 block(256);         // 8 wave32s
  hipLaunchKernelGGL(darts_cell_pool, grid, block, 0, stream, s0, s1, out);
}